// DiffusionConvolution_43121471652032
// MI455X (gfx1250) — compile-verified
//
#include <hip/hip_runtime.h>
#include <hip/hip_bf16.h>

// Problem constants (match reference).
#define NN_DIM 4096   // N
#define FF_DIM 16     // F
#define KK_DIM 3      // diffusion steps

typedef float v2f __attribute__((ext_vector_type(2)));
typedef float v8f __attribute__((ext_vector_type(8)));

// One workgroup = one 16-row tile of out. 8 waves, each reduces a 512-col
// K-chunk with V_WMMA_F32_16X16X4_F32, partials combined through LDS.
__global__ __launch_bounds__(256) void diffusion_wmma_kernel(
    const float* __restrict__ X,      // [N, F]
    const float* __restrict__ theta,  // [K, 2]
    const float* __restrict__ Wp,     // [K, N, N]
    const float* __restrict__ WTp,    // [K, N, N]
    float* __restrict__ out)          // [N, F]
{
  __shared__ float partial[8 * 256];  // 8 waves x (16x16) f32 tile

  const int tid  = threadIdx.x;
  const int wave = tid >> 5;
  const int lane = tid & 31;
  const int n0   = blockIdx.x * 16;   // first output row of this tile
  const int row  = lane & 15;         // A-fragment: M-row within tile
  const int half = lane >> 4;         // A-fragment: K-offset selector (0 -> K0/K1, 1 -> K2/K3)
  const int col  = lane & 15;         // B/C fragment: N column

  // theta is tiny + uniform -> scalar loads.
  const float t00 = theta[0], t01 = theta[1];
  const float t10 = theta[2], t11 = theta[3];
  const float t20 = theta[4], t21 = theta[5];

  const size_t NN = (size_t)NN_DIM * (size_t)NN_DIM;
  const size_t rbase = (size_t)(n0 + row) * (size_t)NN_DIM;

  // Row base pointers for the six streamed matrices (k = 0..2).
  const float* __restrict__ rw0 = Wp  + rbase;
  const float* __restrict__ rw1 = rw0 + NN;
  const float* __restrict__ rw2 = rw1 + NN;
  const float* __restrict__ rt0 = WTp + rbase;
  const float* __restrict__ rt1 = rt0 + NN;
  const float* __restrict__ rt2 = rt1 + NN;

  // This wave's K-chunk: 512 columns -> 128 WMMA steps of K=4.
  const int m_begin = wave * (NN_DIM / 8);
  const int m_end   = m_begin + (NN_DIM / 8);

  v8f c = {};  // 16x16 fp32 accumulator fragment

#pragma unroll 4
  for (int m = m_begin; m < m_end; m += 4) {
    const int o = m + half * 2;  // contiguous pair of columns for this lane

    // Single-use stream -> non-temporal b64 loads (don't thrash L2; X lives there).
    v2f w0 = __builtin_nontemporal_load((const v2f*)(rw0 + o));
    v2f w1 = __builtin_nontemporal_load((const v2f*)(rw1 + o));
    v2f w2 = __builtin_nontemporal_load((const v2f*)(rw2 + o));
    v2f u0 = __builtin_nontemporal_load((const v2f*)(rt0 + o));
    v2f u1 = __builtin_nontemporal_load((const v2f*)(rt1 + o));
    v2f u2 = __builtin_nontemporal_load((const v2f*)(rt2 + o));

    // Build A-fragment of M on the fly: 6-term linear combination (fp32).
    v2f a = t00 * w0 + t10 * w1 + t20 * w2 + t01 * u0 + t11 * u1 + t21 * u2;

    // B-fragment: rows (m + 2*half, m + 2*half + 1) of X, column `col`.
    // X is 256 KB total and read by every wave -> cache-resident.
    const int kr = m + half * 2;
    v2f b;
    b.x = X[(size_t)kr * FF_DIM + col];
    b.y = X[(size_t)(kr + 1) * FF_DIM + col];

    // D = A(16x4, fp32) * B(4x16, fp32) + C  -- full fp32 matrix op.
    c = __builtin_amdgcn_wmma_f32_16x16x4_f32(
        /*neg_a=*/false, a, /*neg_b=*/false, b,
        /*c_mod=*/(short)0, c, /*reuse_a=*/false, /*reuse_b=*/false);
  }

  // Spill this wave's 16x16 partial to LDS in (row, col) order.
#pragma unroll
  for (int j = 0; j < 8; ++j) {
    const int r = j + half * 8;  // C/D layout: VGPR j -> row j (lanes 0-15) / j+8 (lanes 16-31)
    partial[wave * 256 + r * 16 + col] = c[j];
  }
  __syncthreads();

  // Cross-wave reduction + epilogue: out = X + sum of partials.
  {
    const int r = tid >> 4;
    const int cc = tid & 15;
    float s = 0.0f;
#pragma unroll
    for (int w = 0; w < 8; ++w) s += partial[w * 256 + tid];
    const size_t oidx = (size_t)(n0 + r) * FF_DIM + cc;
    out[oidx] = X[oidx] + s;
  }
}

extern "C" void kernel_launch(void* const* d_in, const int* in_sizes, int n_in,
                              void* d_out, int out_size, void* d_ws, size_t ws_size,
                              hipStream_t stream) {
  const float* X     = (const float*)d_in[0];  // [N, F]
  const float* theta = (const float*)d_in[1];  // [K, 2]
  const float* Wp    = (const float*)d_in[2];  // [K, N, N]
  const float* WTp   = (const float*)d_in[3];  // [K, N, N]
  float* out = (float*)d_out;

  dim3 grid(NN_DIM / 16);  // 256 row tiles
  dim3 block(256);         // 8 waves of 32
  diffusion_wmma_kernel<<<grid, block, 0, stream>>>(X, theta, Wp, WTp, out);
}